// TwoStageNet_24773371364018
// MI455X (gfx1250) — compile-verified
//
#include <hip/hip_runtime.h>
#include <hip/hip_bf16.h>
#include <math.h>

typedef float v2f __attribute__((ext_vector_type(2)));
typedef float v8f __attribute__((ext_vector_type(8)));

#define WMMA_F32_4(A, Bm, C) \
    __builtin_amdgcn_wmma_f32_16x16x4_f32(false, (A), false, (Bm), (short)0, (C), false, false)

// ---------------------------------------------------------------------------
// Kernel 1: downscale + conv1(3x3,20ch) + BN + ReLU + 2x2 maxpool + FC1(180->10)
// One wave (32 threads) per block handles 32 samples = two 16-row M-tiles.
// conv1 and FC1 run on v_wmma_f32_16x16x4_f32. All WMMA-feeding LDS loads are
// unconditional (zero-padded B rows make clamped/garbage A reads harmless),
// so EXEC stays all-ones with no saveexec churn in the hot loops.
// ---------------------------------------------------------------------------
__global__ __launch_bounds__(32) void stage1_wmma_kernel(
    const float* __restrict__ x,     // (B,1,28,28)
    const float* __restrict__ w1,    // (20,1,3,3)
    const float* __restrict__ b1,    // (20,)
    const float* __restrict__ g1, const float* __restrict__ be1,
    const float* __restrict__ m1, const float* __restrict__ v1,
    const float* __restrict__ fcw1,  // (10,180)
    const float* __restrict__ fcb1,  // (10,)
    float* __restrict__ logits1,     // (B,10)
    float* __restrict__ xd_ws,       // (B,81) scratch (optional)
    int B, int useWs)
{
    __shared__ float xd_lds[32][81];
    __shared__ float feat_lds[32][184];   // 180 used, padded stride (8B-aligned rows)
    __shared__ float fcw1_lds[16 * 180];  // rows 10..15 zeroed -> unconditional B loads

    const int lane  = threadIdx.x;
    const int half  = lane >> 4;   // 0: lanes 0-15, 1: lanes 16-31
    const int ln    = lane & 15;
    const int sBase = blockIdx.x * 32;
    const int s     = sBase + lane;
    const bool sOK  = (s < B);

    for (int i = lane; i < 16 * 180; i += 32)
        fcw1_lds[i] = (i < 1800) ? fcw1[i] : 0.f;

    // ---- downscale: mean 3x3 over x[:27,:27] -> xd[9][9] (per-lane sample)
    float xd[81];
    #pragma unroll
    for (int i = 0; i < 81; ++i) xd[i] = 0.f;
    if (sOK) {
        const float* xp = x + (size_t)s * 784;
        #pragma unroll
        for (int r = 0; r < 27; ++r) {
            #pragma unroll
            for (int c4 = 0; c4 < 7; ++c4) {
                const float4 v = *reinterpret_cast<const float4*>(xp + r * 28 + c4 * 4);
                const float vv[4] = {v.x, v.y, v.z, v.w};
                #pragma unroll
                for (int e = 0; e < 4; ++e) {
                    const int c = c4 * 4 + e;
                    if (c < 27) xd[(r / 3) * 9 + (c / 3)] += vv[e];
                }
            }
        }
    }
    #pragma unroll
    for (int i = 0; i < 81; ++i) {
        xd[i] *= (1.0f / 9.0f);
        xd_lds[lane][i] = xd[i];
    }
    if (sOK && useWs) {
        #pragma unroll
        for (int i = 0; i < 81; ++i) xd_ws[(size_t)s * 81 + i] = xd[i];
    }
    __syncthreads();

    // ---- conv1 B-fragments (K=9 padded to 12 -> 3 k-steps) and folded BN
    // B 4x16 layout: VGPR0 row = 2*half, VGPR1 row = 2*half+1. Rows k>=9 are 0.
    v2f Bc[2][3];
    float effS[2], effB[2];
    int aoff[3][2];   // per-lane A offsets within a 9x9 xd row (clamped, always < 81)
    #pragma unroll
    for (int ks = 0; ks < 3; ++ks) {
        const int kk = ks * 4 + half * 2;
        const int k1 = kk + 1;
        aoff[ks][0] = (kk / 3) * 9 + (kk % 3);   // kk<=10 -> offset <= 3*9+1, safe
        aoff[ks][1] = (k1 / 3) * 9 + (k1 % 3);   // k1<=11 -> offset <= 3*9+2, safe
    }
    #pragma unroll
    for (int nt = 0; nt < 2; ++nt) {
        const int ch = nt * 16 + ln;
        #pragma unroll
        for (int ks = 0; ks < 3; ++ks) {
            const int kk = ks * 4 + half * 2;
            v2f bv;
            bv.x = (ch < 20 && kk     < 9) ? w1[ch * 9 + kk]     : 0.f;
            bv.y = (ch < 20 && kk + 1 < 9) ? w1[ch * 9 + kk + 1] : 0.f;
            Bc[nt][ks] = bv;
        }
        if (ch < 20) {
            const float sc = g1[ch] * rsqrtf(v1[ch] + 1e-5f);
            effS[nt] = sc;
            effB[nt] = b1[ch] * sc + (be1[ch] - m1[ch] * sc); // conv bias folded into BN
        } else { effS[nt] = 0.f; effB[nt] = 0.f; }
    }

    // ---- conv1 via WMMA, pooling by max over the 4 C-fragments of each 2x2 window
    #pragma unroll
    for (int mt = 0; mt < 2; ++mt) {
        const float* xrow = &xd_lds[mt * 16 + ln][0];   // this lane's A-matrix row
        #pragma unroll
        for (int nt = 0; nt < 2; ++nt) {
            const int ch = nt * 16 + ln;
            for (int ph = 0; ph < 3; ++ph)
            for (int pw = 0; pw < 3; ++pw) {
                v8f vmax;
                #pragma unroll
                for (int v = 0; v < 8; ++v) vmax[v] = -3.0e38f;
                #pragma unroll
                for (int ab = 0; ab < 4; ++ab) {
                    const int boff = (2 * ph + (ab >> 1)) * 9 + (2 * pw + (ab & 1));
                    v8f acc = {};
                    #pragma unroll
                    for (int ks = 0; ks < 3; ++ks) {
                        v2f av;                       // garbage for padded K is * 0
                        av.x = xrow[boff + aoff[ks][0]];
                        av.y = xrow[boff + aoff[ks][1]];
                        acc = WMMA_F32_4(av, Bc[nt][ks], acc);
                    }
                    #pragma unroll
                    for (int v = 0; v < 8; ++v) {
                        const float val = acc[v] * effS[nt] + effB[nt];
                        vmax[v] = fmaxf(vmax[v], val);
                    }
                }
                if (ch < 20) {
                    #pragma unroll
                    for (int v = 0; v < 8; ++v) {
                        const int sm = mt * 16 + v + half * 8;   // C layout: M = v + 8*half
                        feat_lds[sm][ch * 9 + ph * 3 + pw] = fmaxf(vmax[v], 0.f);
                    }
                }
            }
        }
    }
    __syncthreads();

    // ---- FC1 (180 -> 10) via WMMA: 45 k-steps of 16x16x4, b64 LDS loads
    #pragma unroll
    for (int mt = 0; mt < 2; ++mt) {
        const float* frow = &feat_lds[mt * 16 + ln][0];
        const float* wrow = &fcw1_lds[ln * 180];
        v8f acc = {};
        for (int ks = 0; ks < 45; ++ks) {
            const int kk = ks * 4 + half * 2;    // always even -> 8B aligned
            const v2f av = *reinterpret_cast<const v2f*>(frow + kk);
            const v2f bv = *reinterpret_cast<const v2f*>(wrow + kk);
            acc = WMMA_F32_4(av, bv, acc);
        }
        if (ln < 10) {
            const float bias = fcb1[ln];
            #pragma unroll
            for (int v = 0; v < 8; ++v) {
                const int sm = sBase + mt * 16 + v + half * 8;
                if (sm < B) logits1[(size_t)sm * 10 + ln] = acc[v] + bias;
            }
        }
    }
}

// ---------------------------------------------------------------------------
// Kernel 2: log-softmax + top2 -> expert, expert conv (only 4x4 region that
// survives pooling) + BN + ReLU + pool + FC2, scatter/fallback, log-softmax.
// Per-sample expert weights -> VALU path (data-dependent gather of 45 tiny mats).
// ---------------------------------------------------------------------------
__global__ __launch_bounds__(256) void stage2_kernel(
    const float* __restrict__ x,
    const float* __restrict__ logits1,
    const float* __restrict__ xd_ws,
    const float* __restrict__ w2,   // (45,15,25)
    const float* __restrict__ b2,   // (45,15)
    const float* __restrict__ g2, const float* __restrict__ be2,
    const float* __restrict__ m2, const float* __restrict__ v2p,
    const float* __restrict__ fcw2, // (45,2,60)
    const float* __restrict__ fcb2, // (45,2)
    float* __restrict__ out, int B, int useWs)
{
    const int s = blockIdx.x * blockDim.x + threadIdx.x;
    if (s >= B) return;

    float l[10];
    #pragma unroll
    for (int i = 0; i < 10; ++i) l[i] = logits1[(size_t)s * 10 + i];
    float mx = l[0];
    #pragma unroll
    for (int i = 1; i < 10; ++i) mx = fmaxf(mx, l[i]);
    float se = 0.f;
    #pragma unroll
    for (int i = 0; i < 10; ++i) se += expf(l[i] - mx);
    const float lse = logf(se);
    const bool use1 = (-lse) > 0.3f;   // max(log_softmax) == -lse

    // top-2 (>= keeps the highest index among ties, matching stable argsort tail)
    int i1 = 0; float bv1 = -3.0e38f;
    #pragma unroll
    for (int i = 0; i < 10; ++i) if (l[i] >= bv1) { bv1 = l[i]; i1 = i; }
    int i2 = 0; float bv2 = -3.0e38f;
    #pragma unroll
    for (int i = 0; i < 10; ++i) if (i != i1 && l[i] >= bv2) { bv2 = l[i]; i2 = i; }
    const int idx0 = (i1 < i2) ? i1 : i2;
    const int idx1 = (i1 < i2) ? i2 : i1;
    const int eid  = idx0 * (19 - idx0) / 2 + (idx1 - idx0 - 1); // PAIR_TABLE closed form

    // xd: read cached copy from ws (21 MB) instead of re-reading x (205 MB)
    float xd[81];
    if (useWs) {
        #pragma unroll
        for (int i = 0; i < 81; ++i) xd[i] = xd_ws[(size_t)s * 81 + i];
    } else {
        #pragma unroll
        for (int i = 0; i < 81; ++i) xd[i] = 0.f;
        const float* xp = x + (size_t)s * 784;
        #pragma unroll
        for (int r = 0; r < 27; ++r) {
            #pragma unroll
            for (int c4 = 0; c4 < 7; ++c4) {
                const float4 v = *reinterpret_cast<const float4*>(xp + r * 28 + c4 * 4);
                const float vv[4] = {v.x, v.y, v.z, v.w};
                #pragma unroll
                for (int e = 0; e < 4; ++e) {
                    const int c = c4 * 4 + e;
                    if (c < 27) xd[(r / 3) * 9 + (c / 3)] += vv[e];
                }
            }
        }
        #pragma unroll
        for (int i = 0; i < 81; ++i) xd[i] *= (1.0f / 9.0f);
    }

    float s2v0 = fcb2[eid * 2 + 0];
    float s2v1 = fcb2[eid * 2 + 1];
    const float* W   = w2   + (size_t)eid * 375;
    const float* fw0 = fcw2 + (size_t)eid * 120;

    for (int o = 0; o < 15; ++o) {
        float wk[25];
        #pragma unroll
        for (int k = 0; k < 25; ++k) wk[k] = W[o * 25 + k];
        const float sc = g2[eid * 15 + o] * rsqrtf(v2p[eid * 15 + o] + 1e-5f);
        const float sh = be2[eid * 15 + o] - m2[eid * 15 + o] * sc;
        const float bo = b2[eid * 15 + o];

        float hv[16];                       // only :4,:4 survives pooling
        #pragma unroll
        for (int hh = 0; hh < 4; ++hh)
        #pragma unroll
        for (int ww = 0; ww < 4; ++ww) {
            float a = 0.f;
            #pragma unroll
            for (int i = 0; i < 5; ++i)
            #pragma unroll
            for (int j = 0; j < 5; ++j)
                a = fmaf(xd[(i + hh) * 9 + (j + ww)], wk[i * 5 + j], a);
            hv[hh * 4 + ww] = (a + bo) * sc + sh;
        }
        #pragma unroll
        for (int ph = 0; ph < 2; ++ph)
        #pragma unroll
        for (int pw = 0; pw < 2; ++pw) {
            float p = fmaxf(fmaxf(hv[(2 * ph) * 4 + 2 * pw],     hv[(2 * ph) * 4 + 2 * pw + 1]),
                            fmaxf(hv[(2 * ph + 1) * 4 + 2 * pw], hv[(2 * ph + 1) * 4 + 2 * pw + 1]));
            p = fmaxf(p, 0.f);
            const int f = o * 4 + ph * 2 + pw;
            s2v0 = fmaf(fw0[f],      p, s2v0);
            s2v1 = fmaf(fw0[60 + f], p, s2v1);
        }
    }

    float ov[10];
    #pragma unroll
    for (int i = 0; i < 10; ++i) {
        const float v = (i == idx0) ? s2v0 : ((i == idx1) ? s2v1 : -100.f);
        ov[i] = use1 ? l[i] : v;
    }
    float mo = ov[0];
    #pragma unroll
    for (int i = 1; i < 10; ++i) mo = fmaxf(mo, ov[i]);
    float so = 0.f;
    #pragma unroll
    for (int i = 0; i < 10; ++i) so += expf(ov[i] - mo);
    const float lso = logf(so);
    #pragma unroll
    for (int i = 0; i < 10; ++i) out[(size_t)s * 10 + i] = ov[i] - mo - lso;
}

// ---------------------------------------------------------------------------
extern "C" void kernel_launch(void* const* d_in, const int* in_sizes, int n_in,
                              void* d_out, int out_size, void* d_ws, size_t ws_size,
                              hipStream_t stream)
{
    const float* x    = (const float*)d_in[0];
    const float* w1   = (const float*)d_in[1];
    const float* b1   = (const float*)d_in[2];
    const float* g1   = (const float*)d_in[3];
    const float* be1  = (const float*)d_in[4];
    const float* m1   = (const float*)d_in[5];
    const float* v1   = (const float*)d_in[6];
    const float* fcw1 = (const float*)d_in[7];
    const float* fcb1 = (const float*)d_in[8];
    const float* w2   = (const float*)d_in[9];
    const float* b2   = (const float*)d_in[10];
    const float* g2   = (const float*)d_in[11];
    const float* be2  = (const float*)d_in[12];
    const float* m2   = (const float*)d_in[13];
    const float* v2p  = (const float*)d_in[14];
    const float* fcw2 = (const float*)d_in[15];
    const float* fcb2 = (const float*)d_in[16];

    const int B = in_sizes[0] / 784;
    float* logits1 = (float*)d_out;                     // output 0
    float* out2    = (float*)d_out + (size_t)B * 10;    // output 1
    float* xd_ws   = (float*)d_ws;
    const int useWs = (ws_size >= (size_t)B * 81 * sizeof(float)) ? 1 : 0;

    const int nblk1 = (B + 31) / 32;
    stage1_wmma_kernel<<<nblk1, 32, 0, stream>>>(
        x, w1, b1, g1, be1, m1, v1, fcw1, fcb1, logits1, xd_ws, B, useWs);

    const int nblk2 = (B + 255) / 256;
    stage2_kernel<<<nblk2, 256, 0, stream>>>(
        x, logits1, xd_ws, w2, b2, g2, be2, m2, v2p, fcw2, fcb2, out2, B, useWs);
}